// RNNDecoder_29987461660935
// MI455X (gfx1250) — compile-verified
//
#include <hip/hip_runtime.h>
#include <math.h>

typedef float v2f __attribute__((ext_vector_type(2)));
typedef float v8f __attribute__((ext_vector_type(8)));
typedef unsigned int u32x4 __attribute__((ext_vector_type(4)));
typedef int i32x8 __attribute__((ext_vector_type(8)));
typedef int i32x4 __attribute__((ext_vector_type(4)));

#define N_NODES 50000
#define DIM 64
#define D_IN 6
#define N_EDGES 800000
#define NUM_LAYERS 3

#if defined(__HIP_DEVICE_COMPILE__) && __has_builtin(__builtin_amdgcn_tensor_load_to_lds)
#define USE_TDM 1
#else
#define USE_TDM 0
#endif

#if USE_TDM
// ---------------------------------------------------------------------------
// Tensor Data Mover 2-D tile load: global -> LDS, tracked by TENSORcnt.
// D# layout per CDNA5 ISA ch.8:
//   group0: [1:0]=count=1, [63:32]=lds_addr, [120:64]=global_addr, [127:126]=type 2
//   group1: [17:16]=data_size (2 -> 4B), [79:48]=tensor_dim0, [111:80]=tensor_dim1,
//           [127:112]=tile_dim0, [143:128]=tile_dim1, [207:160]=tensor_dim0_stride
//   groups 2/3 (+ extra group on this toolchain): zero (<=2D tile).
// Rows past tensor_dim1 read as zero (free tail padding for the last block).
// This toolchain (clang-23 / therock headers) uses the 6-arg builtin:
//   (u32x4 g0, i32x8 g1, i32x4 g2, i32x4 g3, i32x8 extra, i32 cpol)
// ---------------------------------------------------------------------------
__device__ __forceinline__ void tdm_load_2d(const void* gptr, void* lptr,
                                            unsigned tensor_d0, unsigned tensor_d1,
                                            unsigned tile_d0, unsigned tile_d1,
                                            unsigned stride0) {
  unsigned long long ga = (unsigned long long)(size_t)gptr;
  unsigned lds = (unsigned)(size_t)lptr;  // low 32 bits of generic LDS ptr = LDS offset
  u32x4 g0;
  g0.x = 1u;                                              // count=1, user mode
  g0.y = lds;                                             // lds_addr
  g0.z = (unsigned)ga;                                    // global_addr[31:0]
  g0.w = (unsigned)((ga >> 32) & 0x01FFFFFFu) | (2u << 30);  // addr[56:32] | type=2
  i32x8 g1;
  g1[0] = (int)(2u << 16);                                // data_size=4B, wg_mask=0
  g1[1] = (int)((tensor_d0 & 0xFFFFu) << 16);             // tensor_dim0[15:0]
  g1[2] = (int)(((tensor_d0 >> 16) & 0xFFFFu) | ((tensor_d1 & 0xFFFFu) << 16));
  g1[3] = (int)(((tensor_d1 >> 16) & 0xFFFFu) | ((tile_d0 & 0xFFFFu) << 16));
  g1[4] = (int)(tile_d1 & 0xFFFFu);                       // tile_dim1, tile_dim2=0
  g1[5] = (int)stride0;                                   // tensor_dim0_stride[31:0]
  g1[6] = 0;
  g1[7] = 0;
  i32x4 z4 = {0, 0, 0, 0};
  i32x8 z8 = {0, 0, 0, 0, 0, 0, 0, 0};
  __builtin_amdgcn_tensor_load_to_lds(g0, g1, z4, z4, z8, 0);
}
#endif

// ---------------------------------------------------------------------------
// h = relu(x @ W_nt^T + b_nt)   x:[N,6]  W_nt:[64,6]  -> h:[N,64]
// K=6 is too small for WMMA; pure VALU, memory-bound anyway.
// ---------------------------------------------------------------------------
__global__ __launch_bounds__(256) void nt_kernel(const float* __restrict__ x,
                                                 const float* __restrict__ W,
                                                 const float* __restrict__ b,
                                                 float* __restrict__ h) {
  int tid = blockIdx.x * 256 + threadIdx.x;
  if (tid >= N_NODES * DIM) return;
  int n = tid >> 6, d = tid & 63;
  const float* xr = x + n * D_IN;
  const float* wr = W + d * D_IN;
  float acc = b[d];
#pragma unroll
  for (int k = 0; k < D_IN; ++k) acc = fmaf(xr[k], wr[k], acc);
  h[tid] = fmaxf(acc, 0.0f);
}

// ---------------------------------------------------------------------------
// m = h @ W   h:[N,64]  W:[64,64] row-major (B[k][n] = W[k*64+n])
// 256 threads = 8 waves; each wave computes a 16x64 output tile via
// V_WMMA_F32_16X16X4_F32, K=64 in 16 steps of 4. Tiles staged by the TDM.
// ---------------------------------------------------------------------------
__global__ __launch_bounds__(256) void gemm_kernel(const float* __restrict__ h,
                                                   const float* __restrict__ W,
                                                   float* __restrict__ out) {
  __shared__ float lds_w[64 * 64];    // 16 KB
  __shared__ float lds_h[128 * 64];   // 32 KB
  int tid = threadIdx.x;
  int mblk = blockIdx.x * 128;

#if USE_TDM
  if (tid < 32) {  // one wave issues the DMA; TDM ignores EXEC
    tdm_load_2d(W, lds_w, 4096u, 1u, 4096u, 1u, 4096u);
    tdm_load_2d(h + (size_t)mblk * 64, lds_h, 64u, (unsigned)(N_NODES - mblk),
                64u, 128u, 64u);
    __builtin_amdgcn_s_wait_tensorcnt(0);
  }
  __syncthreads();
#else
  for (int idx = tid; idx < 64 * 64; idx += 256) lds_w[idx] = W[idx];
  for (int idx = tid; idx < 128 * 64; idx += 256) {
    int row = mblk + (idx >> 6);
    lds_h[idx] = (row < N_NODES) ? h[row * 64 + (idx & 63)] : 0.0f;
  }
  __syncthreads();
#endif

  int wave = tid >> 5, lane = tid & 31;
  int half = lane >> 4, r = lane & 15;
  int m0 = wave * 16;
  const float* arow = &lds_h[(m0 + r) * 64];

#pragma unroll
  for (int nt = 0; nt < 4; ++nt) {
    int n0 = nt * 16;
    v8f acc = {};
#pragma unroll
    for (int kk = 0; kk < 16; ++kk) {
      int k = kk * 4 + 2 * half;
      v2f a, bm;
      a.x = arow[k];
      a.y = arow[k + 1];
      bm.x = lds_w[k * 64 + n0 + r];
      bm.y = lds_w[(k + 1) * 64 + n0 + r];
      acc = __builtin_amdgcn_wmma_f32_16x16x4_f32(
          false, a, false, bm, (short)0, acc, false, false);
    }
#pragma unroll
    for (int v = 0; v < 8; ++v) {
      int g = mblk + m0 + v + 8 * half;
      if (g < N_NODES) out[g * 64 + n0 + r] = acc[v];
    }
  }
}

// ---------------------------------------------------------------------------
// agg = 0
// ---------------------------------------------------------------------------
__global__ __launch_bounds__(256) void zero_kernel(float* __restrict__ p, int n) {
  int tid = blockIdx.x * 256 + threadIdx.x;
  if (tid < n) p[tid] = 0.0f;
}

// ---------------------------------------------------------------------------
// agg[dst] += m[src] over all edges.  16 lanes / edge, float4 gather +
// 4 global f32 atomic adds (L2-resident: node working set << 192MB L2).
// ---------------------------------------------------------------------------
__global__ __launch_bounds__(256) void scatter_kernel(const int* __restrict__ ei,
                                                      const float* __restrict__ m,
                                                      float* __restrict__ agg) {
  long long tid = (long long)blockIdx.x * 256 + threadIdx.x;
  if (tid >= (long long)N_EDGES * 16) return;
  int e = (int)(tid >> 4), t = (int)(tid & 15);
  int src = ei[e];
  int dst = ei[N_EDGES + e];
  const float4 val = *(const float4*)(m + (long long)src * 64 + t * 4);
  float* ap = agg + (long long)dst * 64 + t * 4;
  atomicAdd(ap + 0, val.x);
  atomicAdd(ap + 1, val.y);
  atomicAdd(ap + 2, val.z);
  atomicAdd(ap + 3, val.w);
}

// ---------------------------------------------------------------------------
// Fused GRU cell: gi = agg @ W_ih^T + b_ih ; gh = h @ W_hh^T + b_hh
// r = sig(gi_r+gh_r), z = sig(gi_z+gh_z), n = tanh(gi_n + r*gh_n)
// h' = (1-z)*n + z*h
// W_ih/W_hh:[192,64] so B[k][n] = W[n*64+k].  Six WMMA accumulators per
// 16x16 column tile keep the whole gate computation on-register.
// ---------------------------------------------------------------------------
__global__ __launch_bounds__(128) void gru_kernel(const float* __restrict__ agg,
                                                  const float* __restrict__ h,
                                                  const float* __restrict__ W_ih,
                                                  const float* __restrict__ W_hh,
                                                  const float* __restrict__ b_ih,
                                                  const float* __restrict__ b_hh,
                                                  float* __restrict__ hout) {
  __shared__ float lds_a[64 * 64];   // 16 KB
  __shared__ float lds_h[64 * 64];   // 16 KB
  int tid = threadIdx.x;
  int mblk = blockIdx.x * 64;

#if USE_TDM
  if (tid < 32) {
    unsigned rows = (unsigned)(N_NODES - mblk);
    tdm_load_2d(agg + (size_t)mblk * 64, lds_a, 64u, rows, 64u, 64u, 64u);
    tdm_load_2d(h + (size_t)mblk * 64, lds_h, 64u, rows, 64u, 64u, 64u);
    __builtin_amdgcn_s_wait_tensorcnt(0);
  }
  __syncthreads();
#else
  for (int idx = tid; idx < 64 * 64; idx += 128) {
    int row = mblk + (idx >> 6);
    float av = 0.0f, hv = 0.0f;
    if (row < N_NODES) {
      av = agg[row * 64 + (idx & 63)];
      hv = h[row * 64 + (idx & 63)];
    }
    lds_a[idx] = av;
    lds_h[idx] = hv;
  }
  __syncthreads();
#endif

  int wave = tid >> 5, lane = tid & 31;
  int half = lane >> 4, r = lane & 15;
  int m0 = wave * 16;
  const float* arow = &lds_a[(m0 + r) * 64];
  const float* hrow = &lds_h[(m0 + r) * 64];

#pragma unroll
  for (int nt = 0; nt < 4; ++nt) {
    int n0 = nt * 16;
    int col = n0 + r;

    float bir = b_ih[col], biz = b_ih[64 + col], bin = b_ih[128 + col];
    float bhr = b_hh[col], bhz = b_hh[64 + col], bhn = b_hh[128 + col];
    v8f ir = {bir, bir, bir, bir, bir, bir, bir, bir};
    v8f iz = {biz, biz, biz, biz, biz, biz, biz, biz};
    v8f in_ = {bin, bin, bin, bin, bin, bin, bin, bin};
    v8f hr = {bhr, bhr, bhr, bhr, bhr, bhr, bhr, bhr};
    v8f hz = {bhz, bhz, bhz, bhz, bhz, bhz, bhz, bhz};
    v8f hn = {bhn, bhn, bhn, bhn, bhn, bhn, bhn, bhn};

    for (int kk = 0; kk < 16; ++kk) {
      int k = kk * 4 + 2 * half;
      v2f aa, ah;
      aa.x = arow[k];  aa.y = arow[k + 1];
      ah.x = hrow[k];  ah.y = hrow[k + 1];

      v2f wir, wiz, win, whr, whz, whn;
      wir.x = W_ih[col * 64 + k];          wir.y = W_ih[col * 64 + k + 1];
      wiz.x = W_ih[(64 + col) * 64 + k];   wiz.y = W_ih[(64 + col) * 64 + k + 1];
      win.x = W_ih[(128 + col) * 64 + k];  win.y = W_ih[(128 + col) * 64 + k + 1];
      whr.x = W_hh[col * 64 + k];          whr.y = W_hh[col * 64 + k + 1];
      whz.x = W_hh[(64 + col) * 64 + k];   whz.y = W_hh[(64 + col) * 64 + k + 1];
      whn.x = W_hh[(128 + col) * 64 + k];  whn.y = W_hh[(128 + col) * 64 + k + 1];

      ir  = __builtin_amdgcn_wmma_f32_16x16x4_f32(false, aa, false, wir, (short)0, ir,  false, false);
      iz  = __builtin_amdgcn_wmma_f32_16x16x4_f32(false, aa, false, wiz, (short)0, iz,  false, false);
      in_ = __builtin_amdgcn_wmma_f32_16x16x4_f32(false, aa, false, win, (short)0, in_, false, false);
      hr  = __builtin_amdgcn_wmma_f32_16x16x4_f32(false, ah, false, whr, (short)0, hr,  false, false);
      hz  = __builtin_amdgcn_wmma_f32_16x16x4_f32(false, ah, false, whz, (short)0, hz,  false, false);
      hn  = __builtin_amdgcn_wmma_f32_16x16x4_f32(false, ah, false, whn, (short)0, hn,  false, false);
    }

#pragma unroll
    for (int v = 0; v < 8; ++v) {
      int lrow = m0 + v + 8 * half;
      int g = mblk + lrow;
      float rg = 1.0f / (1.0f + expf(-(ir[v] + hr[v])));
      float zg = 1.0f / (1.0f + expf(-(iz[v] + hz[v])));
      float ng = tanhf(in_[v] + rg * hn[v]);
      float hp = lds_h[lrow * 64 + col];
      if (g < N_NODES) hout[g * 64 + col] = (1.0f - zg) * ng + zg * hp;
    }
  }
}

// ---------------------------------------------------------------------------
extern "C" void kernel_launch(void* const* d_in, const int* in_sizes, int n_in,
                              void* d_out, int out_size, void* d_ws, size_t ws_size,
                              hipStream_t stream) {
  const float* x      = (const float*)d_in[0];
  const int*   ei     = (const int*)d_in[1];
  // d_in[2] edge_attr, d_in[3] z_soft: unused by the reference output path.
  const float* W_nt   = (const float*)d_in[4];
  const float* b_nt   = (const float*)d_in[5];
  const float* weight = (const float*)d_in[6];   // [3,64,64]
  const float* W_ih   = (const float*)d_in[7];   // [192,64]
  const float* W_hh   = (const float*)d_in[8];   // [192,64]
  const float* b_ih   = (const float*)d_in[9];
  const float* b_hh   = (const float*)d_in[10];

  const size_t NF = (size_t)N_NODES * DIM;       // 3.2M floats
  float* h0  = (float*)d_ws;
  float* h1  = h0 + NF;
  float* mbf = h1 + NF;
  float* agg = mbf + NF;

  const int nodeElems = N_NODES * DIM;
  dim3 gNT((nodeElems + 255) / 256), b256(256);
  dim3 gGEMM((N_NODES + 127) / 128);
  dim3 gZERO((nodeElems + 255) / 256);
  dim3 gSCAT((N_EDGES * 16 + 255) / 256);
  dim3 gGRU((N_NODES + 63) / 64), b128(128);

  nt_kernel<<<gNT, b256, 0, stream>>>(x, W_nt, b_nt, h0);

  float* hcur = h0;
  float* hnxt = h1;
  for (int layer = 0; layer < NUM_LAYERS; ++layer) {
    float* hdst = (layer == NUM_LAYERS - 1) ? (float*)d_out : hnxt;
    gemm_kernel<<<gGEMM, b256, 0, stream>>>(hcur, weight + (size_t)layer * 64 * 64, mbf);
    zero_kernel<<<gZERO, b256, 0, stream>>>(agg, nodeElems);
    scatter_kernel<<<gSCAT, b256, 0, stream>>>(ei, mbf, agg);
    gru_kernel<<<gGRU, b128, 0, stream>>>(agg, hcur, W_ih, W_hh, b_ih, b_hh, hdst);
    float* t = hcur; hcur = hnxt; hnxt = t;
  }
}